// IndependentMLP_1958505087529
// MI455X (gfx1250) — compile-verified
//
#include <hip/hip_runtime.h>

typedef __attribute__((ext_vector_type(2))) float v2f;
typedef __attribute__((ext_vector_type(4))) float v4f;
typedef __attribute__((ext_vector_type(8))) float v8f;

#define TILES_PER_WAVE 16         // 16-row batch tiles swept by one wave.
                                  // Max 16: store immediate offset (t*512KiB)
                                  // must stay within the signed 24-bit ioffset.
#define TWO_LOG2E 2.8853900817779268f

#if __has_builtin(__builtin_amdgcn_tanhf)
#define NATIVE_TANH 1
#else
#define NATIVE_TANH 0
#endif

#if NATIVE_TANH
// Native V_TANH_F32: single TRANS op, no prescaling anywhere.
#define XSCALE 1.0f
__device__ __forceinline__ float act1(float p) { return __builtin_amdgcn_tanhf(p); }
__device__ __forceinline__ float act2(float a) { return __builtin_amdgcn_tanhf(a); }
#else
// Fallback: tanh(v) = 1 - 2/(exp2(2log2e*v)+1), with scales folded so no
// standalone scale-muls survive in the loop.
#define XSCALE TWO_LOG2E
__device__ __forceinline__ float act1(float p) {   // p = 2log2e * (x*w0)
  float r = __builtin_amdgcn_rcpf(__builtin_amdgcn_exp2f(p) + 1.0f);
  return __builtin_fmaf(r, -2.0f * TWO_LOG2E, TWO_LOG2E);
}
__device__ __forceinline__ float act2(float a) {   // a = 2log2e * pre-act
  float r = __builtin_amdgcn_rcpf(__builtin_amdgcn_exp2f(a) + 1.0f);
  return __builtin_fmaf(r, -2.0f, 1.0f);
}
#endif

// x  : (4096, 1024)        f32
// w0 : (1024, 1, 16)       f32
// w1 : (1024, 16, 16)      f32
// wf : (1024, 16, 8)       f32
// out: (4096, 1024, 1, 8)  f32
__global__ __launch_bounds__(256) void independent_mlp_wmma(
    const float* __restrict__ x,
    const float* __restrict__ w0,
    const float* __restrict__ w1,
    const float* __restrict__ wf,
    float* __restrict__ out)
{
  const int lane = threadIdx.x & 31;
  const int wave = threadIdx.x >> 5;                 // 0..7, one MLP per wave
  const int n    = lane & 15;                        // N index (batch col / M row)
  const int hi   = lane >> 4;                        // lane-half select

  const int i       = ((int)blockIdx.y << 3) + wave;           // MLP 0..1023
  const int b0_base = (int)blockIdx.x * (TILES_PER_WAVE * 16); // first batch row

  const float* __restrict__ w0i = w0 + (size_t)i * 16;
  const float* __restrict__ w1i = w1 + (size_t)i * 256;
  const float* __restrict__ wfi = wf + (size_t)i * 128;

  // ---- Hoisted, loop-invariant weights (all raw loads, nothing to sink) ----
  // Layer 2 transposed: D2 = A2 x B2, A2 = W1^T (M=p chan, K=j), B2 = h1^T.
  //   A layout: lane(16*hi+m): v0=A[m][4c+2hi], v1=A[m][4c+2hi+1]
  //   B layout: lane(16*hi+n): v0=B[4c+2hi][n], v1=B[4c+2hi+1][n]
  // Layer 3 uses a PERMUTED K-slot <-> channel assignment: slot (c,hi,pos)
  // carries channel kp = 8*hi + 2c + pos (a bijection of 0..15), so each lane
  // already owns every channel it must supply for B3 straight from the
  // layer-2 accumulator; A3 loads wf^T rows in the matching permuted order.
  v2f a2[4], a3[4];
  float w0v[8];
#pragma unroll
  for (int c = 0; c < 4; ++c) {
    const int k0 = 4 * c + 2 * hi;                // layer-1 B / layer-2 A slots
    w0v[2 * c]     = w0i[k0];
    w0v[2 * c + 1] = w0i[k0 + 1];
    a2[c][0] = w1i[k0 * 16 + n];                  // W1^T[n][k0]
    a2[c][1] = w1i[(k0 + 1) * 16 + n];
    const int kp = 8 * hi + 2 * c;                // permuted channel for slot
    // Branchless zero-padding of rows m>=8; pinned below so the cndmask
    // cannot be re-materialized inside the tile loop.
    float ax = (n < 8) ? wfi[kp * 8 + (n & 7)]       : 0.0f;
    float ay = (n < 8) ? wfi[(kp + 1) * 8 + (n & 7)] : 0.0f;
    asm volatile("" : "+v"(ax), "+v"(ay));        // keep materialized
    a3[c][0] = ax;
    a3[c][1] = ay;
  }

  // ---- Hoisted base pointers: per-tile accesses become single VMEM ops with
  //      compile-time immediate offsets (24-bit signed ioffset covers all t).
  const float* __restrict__ xp = x + (size_t)(b0_base + n) * 1024 + (size_t)i;
  float* __restrict__ op = out + ((size_t)(b0_base + n) * 1024 + (size_t)i) * 8;

  // ---- Sweep TILES_PER_WAVE batch tiles, prefetching next tile's x ---------
  float xv = xp[0];

#pragma unroll
  for (int t = 0; t < TILES_PER_WAVE; ++t) {
    float xv_next = xv;
    if (t + 1 < TILES_PER_WAVE)                   // compile-time after unroll
      xv_next = xp[(t + 1) * 16 * 1024];          // +64KiB/tile immediate

    // Layer 1 directly in B-matrix layout: h1^T[j][n] = act(x[b0+n]*w0[i][j]).
    const float xs = xv * XSCALE;
    v2f b2[4];
#pragma unroll
    for (int c = 0; c < 4; ++c) {
      b2[c][0] = act1(xs * w0v[2 * c]);
      b2[c][1] = act1(xs * w0v[2 * c + 1]);
    }

    v8f acc = {0.f, 0.f, 0.f, 0.f, 0.f, 0.f, 0.f, 0.f};
#pragma unroll
    for (int c = 0; c < 4; ++c)
      acc = __builtin_amdgcn_wmma_f32_16x16x4_f32(false, a2[c], false, b2[c],
                                                  (short)0, acc, false, false);

    // acc[r] ~ h2pre^T[8*hi + r][n] (pre-scaled in fallback path).
    // B3 slot (c,hi,pos) carries channel 8*hi+2c+pos -> straight from acc.
    v2f b3[4];
#pragma unroll
    for (int c = 0; c < 4; ++c) {
      b3[c][0] = act2(acc[2 * c]);
      b3[c][1] = act2(acc[2 * c + 1]);
    }

    v8f acc3 = {0.f, 0.f, 0.f, 0.f, 0.f, 0.f, 0.f, 0.f};
#pragma unroll
    for (int c = 0; c < 4; ++c)
      acc3 = __builtin_amdgcn_wmma_f32_16x16x4_f32(false, a3[c], false, b3[c],
                                                   (short)0, acc3, false, false);

    // D3 (C layout): lane holds out^T[8*hi + r][n] = out[b0+n][m = 8*hi + r].
    // Valid m = 0..7 -> hi==0 half stores one batch row's 8 contiguous floats
    // as two B128 stores at compile-time immediate offsets (+512KiB/tile).
    if (hi == 0) {
      float* o = op + (size_t)t * 16 * 1024 * 8;
      v4f lo = {acc3[0], acc3[1], acc3[2], acc3[3]};
      v4f hh = {acc3[4], acc3[5], acc3[6], acc3[7]};
      *(v4f*)(o)     = lo;
      *(v4f*)(o + 4) = hh;
    }

    xv = xv_next;
  }
}

extern "C" void kernel_launch(void* const* d_in, const int* in_sizes, int n_in,
                              void* d_out, int out_size, void* d_ws, size_t ws_size,
                              hipStream_t stream) {
  const float* x  = (const float*)d_in[0];
  const float* w0 = (const float*)d_in[1];
  const float* w1 = (const float*)d_in[2];
  const float* wf = (const float*)d_in[3];
  float* out      = (float*)d_out;

  // grid.x: 16 groups of (TILES_PER_WAVE=16) batch tiles (4096 rows / 256)
  // grid.y: 128 groups of 8 MLPs (one MLP per wave)
  dim3 grid(16, 128), block(256);
  hipLaunchKernelGGL(independent_mlp_wmma, grid, block, 0, stream,
                     x, w0, w1, wf, out);
}